// RNN_11364483465931
// MI455X (gfx1250) — compile-verified
//
#include <hip/hip_runtime.h>

typedef __attribute__((ext_vector_type(2))) float v2f;
typedef __attribute__((ext_vector_type(8))) float v8f;

#define SEQ   512
#define BATCH 1024
#define EMB   100
#define HID   6
#define D_FF  200
#define D_OUT 2

// ---------------------------------------------------------------------------
// Kernel 1: pre0[m][h] = x[m][:] . W_ih0[h][:] + (b_ih0[h]+b_hh0[h])
//   m = s*BATCH + b, M = 524288 rows. One wave = one 16x16 tile via
//   V_WMMA_F32_16X16X4_F32 (25 K-steps of 4).
// A tiles staged to LDS with async global->LDS b64 copies (ASYNCcnt),
// B matrix staged zero-padded in LDS so the inner loop is branch-free.
// ---------------------------------------------------------------------------
__global__ __launch_bounds__(256)
void pre0_wmma_kernel(const float* __restrict__ x,
                      const float* __restrict__ W_ih0,
                      const float* __restrict__ b_ih0,
                      const float* __restrict__ b_hh0,
                      float* __restrict__ pre0)
{
    // [0, 1600)            : Wpad, 16 x 100, rows >= 6 zeroed
    // [1600 + w*1600, ...) : per-wave A tile, 16 x 100
    __shared__ float smem[16 * EMB + 8 * 16 * EMB];

    const int tid  = threadIdx.x;
    const int lane = tid & 31;
    const int wave = tid >> 5;

    // ---- stage zero-padded W (W_ih0 is [6][100] contiguous -> first 600) ----
    for (int i = tid; i < 16 * EMB; i += 256) {
        smem[i] = (i < HID * EMB) ? W_ih0[i] : 0.0f;
    }

    // ---- async-copy this wave's A tile (1600 contiguous floats of x) -------
    const long  m0    = ((long)blockIdx.x * 8 + wave) * 16;
    float*      atile = smem + 16 * EMB + wave * 16 * EMB;
    const unsigned atile_lds = (unsigned)(uintptr_t)atile;          // LDS byte addr
    const unsigned long long gtile = (unsigned long long)(uintptr_t)(x + (size_t)m0 * EMB);

#pragma unroll
    for (int it = 0; it < 25; ++it) {                               // 25 * 32 lanes * 8B = 6400B
        const unsigned byteOff = (unsigned)((lane + 32 * it) * 8);
        unsigned           ldsAddr = atile_lds + byteOff;
        unsigned long long gAddr   = gtile + byteOff;
        asm volatile("global_load_async_to_lds_b64 %0, %1, off"
                     :: "v"(ldsAddr), "v"(gAddr)
                     : "memory");
    }

    __syncthreads();                                   // Wpad visible to all waves
    asm volatile("s_wait_asynccnt 0x0" ::: "memory");  // A tile landed in LDS

    // ---- WMMA main loop ----------------------------------------------------
    const int n     = lane & 15;              // output column / B row select
    const int khalf = (lane >> 4) << 1;       // 0 (lanes 0-15) or 2 (lanes 16-31)

    const float bias = (n < HID) ? (b_ih0[n] + b_hh0[n]) : 0.0f;
    v8f c;
#pragma unroll
    for (int j = 0; j < 8; ++j) c[j] = bias;

    const float* __restrict__ arow = atile + (lane & 15) * EMB + khalf;
    const float* __restrict__ brow = smem  + n * EMB + khalf;   // zero-padded, always valid

#pragma unroll
    for (int k = 0; k < EMB; k += 4) {        // 25 WMMA steps, branch-free
        v2f a  = *(const v2f*)(arow + k);     // ds_load_b64 (8B aligned)
        v2f bb = *(const v2f*)(brow + k);     // ds_load_b64 (8B aligned)
        c = __builtin_amdgcn_wmma_f32_16x16x4_f32(
                false, a, false, bb, (short)0, c, false, false);
    }

    // D layout: VGPR j -> M = j (lanes 0-15) or 8+j (lanes 16-31), N = lane&15
    if (n < HID) {
        const long mbase = m0 + ((lane >> 4) << 3);
#pragma unroll
        for (int j = 0; j < 8; ++j) {
            pre0[(size_t)(mbase + j) * HID + n] = c[j];
        }
    }
}

// ---------------------------------------------------------------------------
// Kernel 2: sequential 512-step 2-layer RNN scan (one thread per batch
// element, 6x6 weights in registers) + fused MLP head (6->200->2, no relu).
// ---------------------------------------------------------------------------
__global__ __launch_bounds__(256)
void rnn_scan_head_kernel(const float* __restrict__ pre0,
                          const float* __restrict__ W_hh0,
                          const float* __restrict__ W_ih1,
                          const float* __restrict__ W_hh1,
                          const float* __restrict__ b_ih1,
                          const float* __restrict__ b_hh1,
                          const float* __restrict__ W1,
                          const float* __restrict__ b1,
                          const float* __restrict__ W2,
                          const float* __restrict__ b2,
                          float* __restrict__ out)
{
    const int b = blockIdx.x * blockDim.x + threadIdx.x;   // 0..1023

    float whh0[HID * HID], wih1[HID * HID], whh1[HID * HID], bias1[HID];
#pragma unroll
    for (int i = 0; i < HID * HID; ++i) {
        whh0[i] = W_hh0[i];
        wih1[i] = W_ih1[i];
        whh1[i] = W_hh1[i];
    }
#pragma unroll
    for (int i = 0; i < HID; ++i) bias1[i] = b_ih1[i] + b_hh1[i];

    float h0[HID], h1[HID];
#pragma unroll
    for (int i = 0; i < HID; ++i) { h0[i] = 0.0f; h1[i] = 0.0f; }

    for (int t = 0; t < SEQ; ++t) {
        const float* __restrict__ p = pre0 + ((size_t)t * BATCH + b) * HID;
        float p0[HID];
#pragma unroll
        for (int i = 0; i < HID; ++i) p0[i] = p[i];

        float h0n[HID];
#pragma unroll
        for (int i = 0; i < HID; ++i) {
            float acc = p0[i];
#pragma unroll
            for (int j = 0; j < HID; ++j) acc = fmaf(h0[j], whh0[i * HID + j], acc);
            h0n[i] = fmaxf(acc, 0.0f);
        }

        float h1n[HID];
#pragma unroll
        for (int i = 0; i < HID; ++i) {
            float acc = bias1[i];
#pragma unroll
            for (int j = 0; j < HID; ++j) acc = fmaf(h0n[j], wih1[i * HID + j], acc);
#pragma unroll
            for (int j = 0; j < HID; ++j) acc = fmaf(h1[j], whh1[i * HID + j], acc);
            h1n[i] = fmaxf(acc, 0.0f);
        }

#pragma unroll
        for (int i = 0; i < HID; ++i) { h0[i] = h0n[i]; h1[i] = h1n[i]; }
    }

    // MLP head: out = (h1 @ W1^T + b1) @ W2^T + b2   (no activation)
    float o0 = b2[0];
    float o1 = b2[1];
#pragma unroll 4
    for (int f = 0; f < D_FF; ++f) {
        float hf = b1[f];
#pragma unroll
        for (int k = 0; k < HID; ++k) hf = fmaf(h1[k], W1[f * HID + k], hf);
        o0 = fmaf(hf, W2[f], o0);
        o1 = fmaf(hf, W2[D_FF + f], o1);
    }
    out[b * 2 + 0] = o0;
    out[b * 2 + 1] = o1;
}

// ---------------------------------------------------------------------------
extern "C" void kernel_launch(void* const* d_in, const int* in_sizes, int n_in,
                              void* d_out, int out_size, void* d_ws, size_t ws_size,
                              hipStream_t stream)
{
    const float* x     = (const float*)d_in[0];
    const float* W_ih0 = (const float*)d_in[1];
    const float* W_hh0 = (const float*)d_in[2];
    const float* b_ih0 = (const float*)d_in[3];
    const float* b_hh0 = (const float*)d_in[4];
    const float* W_ih1 = (const float*)d_in[5];
    const float* W_hh1 = (const float*)d_in[6];
    const float* b_ih1 = (const float*)d_in[7];
    const float* b_hh1 = (const float*)d_in[8];
    const float* W1    = (const float*)d_in[9];
    const float* b1    = (const float*)d_in[10];
    const float* W2    = (const float*)d_in[11];
    const float* b2    = (const float*)d_in[12];

    float* pre0 = (float*)d_ws;   // SEQ*BATCH*HID floats = 12.6 MB

    // (SEQ*BATCH)/16 = 32768 M-tiles, 8 waves (tiles) per 256-thread block
    pre0_wmma_kernel<<<dim3(4096), dim3(256), 0, stream>>>(x, W_ih0, b_ih0, b_hh0, pre0);

    // one thread per batch element
    rnn_scan_head_kernel<<<dim3(BATCH / 256), dim3(256), 0, stream>>>(
        pre0, W_hh0, W_ih1, W_hh1, b_ih1, b_hh1, W1, b1, W2, b2, (float*)d_out);
}